// IterNorm_3899830304787
// MI455X (gfx1250) — compile-verified
//
#include <hip/hip_runtime.h>
#include <hip/hip_bf16.h>

// ---------------- problem constants ----------------
#define NB    64                 // batch
#define NCH   256                // channels
#define HW    3136               // 56*56
#define CHW   (NCH * HW)         // 802816
#define MTOT  (NB * HW)          // 200704 samples per channel
#define NG    4                  // groups
#define NCg   64                 // channels per group
#define EPSV  1e-5f
#define TITER 5

#define TS    64                 // k-tile size (divides HW -> never crosses batch)
#define LSTR  66                 // padded LDS row stride (floats) to dodge bank conflicts
#define CPW   16                 // k-chunks per workgroup
#define KBLK  (MTOT / TS)        // 3136 chunks per group
#define WGPG  (KBLK / CPW)       // 196 workgroups per group

typedef __attribute__((ext_vector_type(2))) float        v2f;
typedef __attribute__((ext_vector_type(8))) float        v8f;
typedef __attribute__((ext_vector_type(4))) unsigned int u32x4;
typedef __attribute__((ext_vector_type(8))) int          i32x8;
typedef __attribute__((ext_vector_type(4))) int          i32x4;

// A-fragment (and symmetric-Gram B-fragment): matrix stored row-major [M][K] in LDS.
// lane l: M = l&15, VGPR0/1 = M-row at K = kbase + 2*(l>>4) + {0,1}
__device__ __forceinline__ v2f frag_row(const float* lds, int rowblk, int kbase) {
    int l  = threadIdx.x & 31;
    int ch = rowblk + (l & 15);
    int kk = kbase + 2 * (l >> 4);
    const float* p = lds + ch * LSTR + kk;
    return v2f{p[0], p[1]};
}

// B-fragment: matrix stored row-major [K][N] in LDS.
// lane l: N = l&15, VGPR0/1 = rows K = kbase + 2*(l>>4) + {0,1} at column N
__device__ __forceinline__ v2f frag_col(const float* lds, int colblk, int kbase) {
    int l  = threadIdx.x & 31;
    int n  = colblk + (l & 15);
    int kk = kbase + 2 * (l >> 4);
    return v2f{lds[kk * LSTR + n], lds[(kk + 1) * LSTR + n]};
}

// ---- Tensor Data Mover: DMA a 64(ch) x 64(k) fp32 tile, row stride HW floats,
// into LDS with 2-DWORD pad every 64 DWORDs (-> LSTR = 66).  D# per ISA ch.8.
// clang-23 toolchain: 6-arg builtin (u32x4, i32x8, i32x4, i32x4, i32x8, i32).
__device__ __forceinline__ void tdm_load_tile(unsigned int lds_off, const float* gptr) {
    unsigned long long ga = (unsigned long long)(uintptr_t)gptr;   // 57-bit byte addr
    u32x4 g0 = {
        1u,                                        // count=1 (valid user descriptor)
        lds_off,                                   // LDS byte address (tile start)
        (unsigned int)ga,                          // global_addr[31:0]
        (unsigned int)(ga >> 32) | (2u << 30)      // global_addr[56:32] | type=2
    };
    i32x8 g1 = {
        (int)((2u << 16) |                         // data_size = 4 bytes
              (1u << 20) |                         // pad_enable
              (5u << 22) |                         // pad_interval: every 64 DWORDs
              (1u << 25)),                         // pad_amount: 2 DWORDs
        (int)(64u << 16),                          // [15:0] atomic_bar=0 | tensor_dim0.lo=64
        (int)(64u << 16),                          // tensor_dim0.hi=0 | tensor_dim1.lo=64
        (int)(64u << 16),                          // tensor_dim1.hi=0 | tile_dim0=64
        (int)(64u),                                // tile_dim1=64 | tile_dim2=0
        (int)HW,                                   // tensor_dim0_stride.lo = 3136 elems
        0,                                         // stride0.hi | stride1.lo
        0                                          // stride1.hi
    };
    i32x4 z4 = {0, 0, 0, 0};
    i32x8 z8 = {0, 0, 0, 0, 0, 0, 0, 0};
    __builtin_amdgcn_tensor_load_to_lds(g0, g1, z4, z4, z8, 0);
}

// ---------------- kernel 0: zero stats workspace ----------------
__global__ void k0_zero(float* __restrict__ p, int n) {
    int i = blockIdx.x * blockDim.x + threadIdx.x;
    if (i < n) p[i] = 0.0f;
}

// ---------------- kernel 1: channel sums + group Gram via WMMA ----------------
__global__ __launch_bounds__(256) void k1_stats(const float* __restrict__ x,
                                                float* __restrict__ sums,
                                                float* __restrict__ gram) {
    __shared__ __align__(16) float tile[NCg * LSTR];

    const int gi = blockIdx.x / WGPG;
    const int kb = blockIdx.x % WGPG;
    const int t  = threadIdx.x;
    const int l  = t & 31;
    const int wv = t >> 5;
    const int r0 = t >> 4;           // 0..15 : load-row base
    const int c4 = (t & 15) * 4;     // 0..60 : load-col (float4)

    const int t0 = wv * 2, t1 = wv * 2 + 1;
    const int tr0 = t0 >> 2, tc0 = t0 & 3;
    const int tr1 = t1 >> 2, tc1 = t1 & 3;

    float lsum[4] = {0.f, 0.f, 0.f, 0.f};
    v8f acc0 = {};
    v8f acc1 = {};

    const float* xg = x + (size_t)gi * NCg * HW;

    for (int c = 0; c < CPW; ++c) {
        const int k0 = (kb * CPW + c) * TS;
        const int b  = k0 / HW;
        const int r  = k0 % HW;
        const float* base = xg + (size_t)b * CHW + r;

        // coalesced 64ch x 64k tile load (b128), fused per-channel sum
        #pragma unroll
        for (int i = 0; i < 4; ++i) {
            const int row = r0 + 16 * i;
            const float4 v = *(const float4*)(base + (size_t)row * HW + c4);
            *(float4*)(&tile[row * LSTR + c4]) = v;
            lsum[i] += v.x + v.y + v.z + v.w;
        }
        __syncthreads();

        // Gram accumulation: D += A * A^T, fp32 WMMA 16x16x4, K = 64 in 16 steps
        #pragma unroll
        for (int s = 0; s < 16; ++s) {
            v2f a0 = frag_row(tile, tr0 * 16, s * 4);
            v2f b0 = frag_row(tile, tc0 * 16, s * 4);   // B[k][j] = X[j][k] -> same loader
            acc0 = __builtin_amdgcn_wmma_f32_16x16x4_f32(false, a0, false, b0,
                                                         (short)0, acc0, false, false);
            v2f a1 = frag_row(tile, tr1 * 16, s * 4);
            v2f b1 = frag_row(tile, tc1 * 16, s * 4);
            acc1 = __builtin_amdgcn_wmma_f32_16x16x4_f32(false, a1, false, b1,
                                                         (short)0, acc1, false, false);
        }
        __syncthreads();
    }

    // reduce partial Gram tiles into global workspace
    float* gg = gram + gi * (NCg * NCg);
    const int hi  = (l >> 4) * 8;
    const int col = l & 15;
    #pragma unroll
    for (int v = 0; v < 8; ++v) {
        atomicAdd(&gg[(tr0 * 16 + v + hi) * NCg + tc0 * 16 + col], acc0[v]);
        atomicAdd(&gg[(tr1 * 16 + v + hi) * NCg + tc1 * 16 + col], acc1[v]);
    }
    #pragma unroll
    for (int i = 0; i < 4; ++i)
        atomicAdd(&sums[gi * NCg + r0 + 16 * i], lsum[i]);
}

// ---------------- kernel 2: Newton-Schulz on 64x64, one WG per group ----------------
// 64x64x64 matmul, 8 waves x 2 tiles. Internal barrier lets D alias A/B.
__device__ __forceinline__ void mm64(float* D, const float* A, const float* B) {
    const int t  = threadIdx.x;
    const int wv = t >> 5;
    const int l  = t & 31;
    v8f acc[2] = {{}, {}};
    #pragma unroll
    for (int q = 0; q < 2; ++q) {
        const int ti = wv * 2 + q, tr = ti >> 2, tc = ti & 3;
        #pragma unroll
        for (int s = 0; s < 16; ++s) {
            v2f a = frag_row(A, tr * 16, s * 4);
            v2f b = frag_col(B, tc * 16, s * 4);
            acc[q] = __builtin_amdgcn_wmma_f32_16x16x4_f32(false, a, false, b,
                                                           (short)0, acc[q], false, false);
        }
    }
    __syncthreads();   // all reads of A/B done -> safe even if D aliases them
    const int hi = (l >> 4) * 8, col = l & 15;
    #pragma unroll
    for (int q = 0; q < 2; ++q) {
        const int ti = wv * 2 + q, tr = ti >> 2, tc = ti & 3;
        #pragma unroll
        for (int v = 0; v < 8; ++v)
            D[(tr * 16 + v + hi) * LSTR + tc * 16 + col] = acc[q][v];
    }
}

__global__ __launch_bounds__(256) void k2_ns(const float* __restrict__ sums,
                                             const float* __restrict__ gram,
                                             const float* __restrict__ weight,
                                             const float* __restrict__ bias,
                                             float* __restrict__ wmw,
                                             float* __restrict__ betaw) {
    __shared__ __align__(16) float Sn[NCg * LSTR];
    __shared__ __align__(16) float P [NCg * LSTR];
    __shared__ __align__(16) float T1[NCg * LSTR];
    __shared__ float mean[NCg];
    __shared__ float rtr_s;

    const int gi = blockIdx.x;
    const int t  = threadIdx.x;

    if (t < NCg) mean[t] = sums[gi * NCg + t] * (1.0f / (float)MTOT);
    __syncthreads();

    // Sigma = G/m - mu mu^T + eps I
    for (int e = t; e < NCg * NCg; e += 256) {
        const int i = e >> 6, j = e & 63;
        float s = gram[gi * NCg * NCg + e] * (1.0f / (float)MTOT)
                - mean[i] * mean[j] + ((i == j) ? EPSV : 0.0f);
        Sn[i * LSTR + j] = s;
    }
    __syncthreads();
    if (t == 0) {
        float tr = 0.0f;
        for (int i = 0; i < NCg; ++i) tr += Sn[i * LSTR + i];
        rtr_s = 1.0f / tr;
    }
    __syncthreads();
    const float rtr = rtr_s;

    for (int e = t; e < NCg * NCg; e += 256) {
        const int i = e >> 6, j = e & 63;
        Sn[i * LSTR + j] *= rtr;
        P [i * LSTR + j] = (i == j) ? 1.0f : 0.0f;
    }
    __syncthreads();

    for (int it = 0; it < TITER; ++it) {
        mm64(T1, P, P);    __syncthreads();   // T1 = P^2
        mm64(T1, T1, P);   __syncthreads();   // T1 = P^3   (alias-safe)
        mm64(T1, T1, Sn);  __syncthreads();   // T1 = P^3 Sn
        for (int e = t; e < NCg * NCg; e += 256) {
            const int i = e >> 6, j = e & 63;
            P[i * LSTR + j] = 1.5f * P[i * LSTR + j] - 0.5f * T1[i * LSTR + j];
        }
        __syncthreads();
    }

    const float wsc = sqrtf(rtr);
    for (int e = t; e < NCg * NCg; e += 256) {
        const int i = e >> 6, j = e & 63;
        wmw[gi * NCg * NCg + e] = P[i * LSTR + j] * wsc;
    }
    if (t < NCg) {  // beta_c = bias_c - (wm mu)_c * weight_c
        float cm = 0.0f;
        for (int j = 0; j < NCg; ++j) cm += P[t * LSTR + j] * wsc * mean[j];
        const int cg = gi * NCg + t;
        betaw[cg] = bias[cg] - cm * weight[cg];
    }
}

// ---------------- kernel 3: apply whitening, out = wm@x * w + beta ----------------
// TDM double-buffered X staging: wave 0 issues tensor_load_to_lds for chunk c+1,
// waits TENSORcnt<=1 (chunk c complete, in-order), all waves WMMA on chunk c.
__global__ __launch_bounds__(256) void k3_apply(const float* __restrict__ x,
                                                const float* __restrict__ wmw,
                                                const float* __restrict__ betaw,
                                                const float* __restrict__ weight,
                                                float* __restrict__ out) {
    __shared__ __align__(16) float Wm[NCg * LSTR];
    __shared__ __align__(16) float Xs[2][NCg * LSTR];
    __shared__ float wfac[NCg], bfac[NCg];

    const int gi = blockIdx.x / WGPG;
    const int kb = blockIdx.x % WGPG;
    const int t  = threadIdx.x;
    const int l  = t & 31;
    const int wv = t >> 5;

    const float* xg = x   + (size_t)gi * NCg * HW;
    float*       og = out + (size_t)gi * NCg * HW;

    const unsigned int xs_off[2] = {
        (unsigned int)(uintptr_t)(void*)&Xs[0][0],   // flat-LDS low 32 bits = LDS offset
        (unsigned int)(uintptr_t)(void*)&Xs[1][0]
    };

    // prologue: kick TDM for chunk 0 while the block stages wm / scale factors
    if (wv == 0) {
        const int k0 = (kb * CPW) * TS;
        tdm_load_tile(xs_off[0], xg + (size_t)(k0 / HW) * CHW + (k0 % HW));
    }
    for (int e = t; e < NCg * NCg; e += 256)
        Wm[(e >> 6) * LSTR + (e & 63)] = wmw[gi * NCg * NCg + e];
    if (t < NCg) {
        wfac[t] = weight[gi * NCg + t];
        bfac[t] = betaw[gi * NCg + t];
    }
    __syncthreads();

    const int ci  = wv >> 1;          // channel tile-row 0..3
    const int sj0 = (wv & 1) * 2;     // spatial tile-cols {0,1} or {2,3}

    // preload A fragments of wm for this wave's channel block; reused every chunk
    v2f afrag[16];
    #pragma unroll
    for (int s = 0; s < 16; ++s) afrag[s] = frag_row(Wm, ci * 16, s * 4);

    const int hi = (l >> 4) * 8, col = l & 15;

    for (int c = 0; c < CPW; ++c) {
        if (wv == 0) {
            if (c + 1 < CPW) {       // issue next chunk's DMA, then wait for current
                const int kn = (kb * CPW + c + 1) * TS;
                tdm_load_tile(xs_off[(c + 1) & 1],
                              xg + (size_t)(kn / HW) * CHW + (kn % HW));
                __builtin_amdgcn_s_wait_tensorcnt(1);
            } else {
                __builtin_amdgcn_s_wait_tensorcnt(0);
            }
        }
        __syncthreads();             // chunk c resident for all waves

        const float* Xcur = &Xs[c & 1][0];
        const int k0 = (kb * CPW + c) * TS;
        float* obase = og + (size_t)(k0 / HW) * CHW + (k0 % HW);

        #pragma unroll
        for (int q = 0; q < 2; ++q) {
            const int sj = sj0 + q;
            v8f acc = {};
            #pragma unroll
            for (int s = 0; s < 16; ++s) {
                v2f bfr = frag_col(Xcur, sj * 16, s * 4);   // B[k=ch][n=spatial]
                acc = __builtin_amdgcn_wmma_f32_16x16x4_f32(false, afrag[s], false, bfr,
                                                            (short)0, acc, false, false);
            }
            // direct store: half-wave writes a contiguous 64B row segment
            #pragma unroll
            for (int v = 0; v < 8; ++v) {
                const int ch = ci * 16 + v + hi;
                obase[(size_t)ch * HW + sj * 16 + col] = acc[v] * wfac[ch] + bfac[ch];
            }
        }
        __syncthreads();             // all reads of Xs[c&1] done before TDM reuses it
    }
}

// ---------------- launch ----------------
extern "C" void kernel_launch(void* const* d_in, const int* in_sizes, int n_in,
                              void* d_out, int out_size, void* d_ws, size_t ws_size,
                              hipStream_t stream) {
    const float* x      = (const float*)d_in[0];
    const float* weight = (const float*)d_in[1];
    const float* bias   = (const float*)d_in[2];
    float* out = (float*)d_out;
    float* ws  = (float*)d_ws;

    float* sums  = ws;                 // 256
    float* gram  = ws + 256;           // 4*4096 = 16384
    float* wmw   = ws + 16640;         // 4*4096 = 16384
    float* betaw = ws + 33024;         // 256
    // total ws use: 33280 floats (~130 KB)

    const int nz = 16640;              // sums + gram must start at zero (ws is poisoned)
    k0_zero<<<(nz + 255) / 256, 256, 0, stream>>>(ws, nz);
    k1_stats<<<NG * WGPG, 256, 0, stream>>>(x, sums, gram);
    k2_ns<<<NG, 256, 0, stream>>>(sums, gram, weight, bias, wmw, betaw);
    k3_apply<<<NG * WGPG, 256, 0, stream>>>(x, wmw, betaw, weight, out);
}